// Attn_81690277970335
// MI455X (gfx1250) — compile-verified
//
#include <hip/hip_runtime.h>
#include <hip/hip_bf16.h>

#define HIDDEN 1024
#define SEQ    32768

typedef __attribute__((ext_vector_type(2))) float v2f;
typedef __attribute__((ext_vector_type(8))) float v8f;

// ---------------------------------------------------------------------------
// Phase A0: partial matvec  partial[chunk][col] = sum_{o in chunk} W[o][col]*h[o]
// grid = (4 col-groups of 256, 8 row-chunks of 128), block = 256
// ---------------------------------------------------------------------------
__global__ void matvec_partial(const float* __restrict__ W,
                               const float* __restrict__ hidden,
                               float* __restrict__ partial) {
    __shared__ float h_lds[128];
    const int col  = blockIdx.x * 256 + threadIdx.x;
    const int row0 = blockIdx.y * 128;
    if (threadIdx.x < 128) h_lds[threadIdx.x] = hidden[row0 + threadIdx.x];
    __syncthreads();
    float acc = 0.0f;
#pragma unroll 8
    for (int o = 0; o < 128; ++o)
        acc = fmaf(W[(size_t)(row0 + o) * HIDDEN + col], h_lds[o], acc);
    partial[blockIdx.y * HIDDEN + col] = acc;
}

// Phase A1: v[col] = sum over 8 row-chunk partials.  grid = 4, block = 256
__global__ void matvec_reduce(const float* __restrict__ partial,
                              float* __restrict__ v) {
    const int col = blockIdx.x * 256 + threadIdx.x;
    float s = 0.0f;
#pragma unroll
    for (int j = 0; j < 8; ++j) s += partial[j * HIDDEN + col];
    v[col] = s;
}

// ---------------------------------------------------------------------------
// Phase B: energies[s] = enc[s,:] . v   via V_WMMA_F32_16X16X4_F32.
// block = 256 (8 waves), each wave computes 16 rows; grid = SEQ/128 = 256.
// A fragment (16x4 f32): lane L (m=L&15, half=L>>4) holds enc[row m][k+2*half .. +1]
// B fragment: all 16 columns = v[k..k+3]  => every column of D is the dot product.
// D layout: VGPR j, lanes 0-15 -> M=j, lanes 16-31 -> M=8+j.
// ---------------------------------------------------------------------------
__global__ void energies_wmma(const float* __restrict__ enc,
                              const float* __restrict__ v,
                              float* __restrict__ e) {
    __shared__ float vlds[HIDDEN];
    for (int i = threadIdx.x; i < HIDDEN; i += 256) vlds[i] = v[i];
    __syncthreads();

    const int lane    = threadIdx.x & 31;
    const int wave    = threadIdx.x >> 5;
    const int half    = lane >> 4;      // 0 => K 0..1, 1 => K 2..3
    const int m       = lane & 15;
    const int rowBase = blockIdx.x * 128 + wave * 16;
    const float* __restrict__ arow = enc + (size_t)(rowBase + m) * HIDDEN;

    v8f acc = {};
#pragma unroll 4
    for (int k = 0; k < HIDDEN; k += 4) {
        const int kk = k + 2 * half;
        v2f a = *(const v2f*)(arow + kk);   // global_load_b64, fills A VGPR pair
        v2f b = *(const v2f*)(vlds + kk);   // ds_load_b64 (broadcast within half)
        acc = __builtin_amdgcn_wmma_f32_16x16x4_f32(
            /*neg_a=*/false, a, /*neg_b=*/false, b,
            /*c_mod=*/(short)0, acc, /*reuse_a=*/false, /*reuse_b=*/false);
    }
    if (m == 0) {  // lane 0 holds rows 0..7 (col 0); lane 16 holds rows 8..15
#pragma unroll
        for (int j = 0; j < 8; ++j)
            e[rowBase + half * 8 + j] = acc[j];
    }
}

// ---------------------------------------------------------------------------
// Phase C1: per-block (512 elems) max + sum(exp(x - max)).  grid=64, block=256
// ---------------------------------------------------------------------------
__global__ void softmax_part(const float* __restrict__ e,
                             float* __restrict__ stats) {
    __shared__ float red[256];
    const int base = blockIdx.x * 512;
    const float x0 = e[base + threadIdx.x];
    const float x1 = e[base + 256 + threadIdx.x];
    red[threadIdx.x] = fmaxf(x0, x1);
    __syncthreads();
    for (int s = 128; s > 0; s >>= 1) {
        if (threadIdx.x < s)
            red[threadIdx.x] = fmaxf(red[threadIdx.x], red[threadIdx.x + s]);
        __syncthreads();
    }
    const float bm = red[0];
    __syncthreads();
    red[threadIdx.x] = __expf(x0 - bm) + __expf(x1 - bm);
    __syncthreads();
    for (int s = 128; s > 0; s >>= 1) {
        if (threadIdx.x < s)
            red[threadIdx.x] += red[threadIdx.x + s];
        __syncthreads();
    }
    if (threadIdx.x == 0) {
        stats[2 * blockIdx.x]     = bm;
        stats[2 * blockIdx.x + 1] = red[0];
    }
}

// Phase C2: combine 64 (max,sum) pairs -> global max + 1/sum.  1 block of 64
__global__ void softmax_combine(const float* __restrict__ stats,
                                float* __restrict__ gm) {
    __shared__ float mred[64];
    __shared__ float sred[64];
    const int t = threadIdx.x;
    const float m = stats[2 * t];
    const float s = stats[2 * t + 1];
    mred[t] = m;
    __syncthreads();
    for (int k = 32; k > 0; k >>= 1) {
        if (t < k) mred[t] = fmaxf(mred[t], mred[t + k]);
        __syncthreads();
    }
    const float g = mred[0];
    __syncthreads();
    sred[t] = s * __expf(m - g);
    __syncthreads();
    for (int k = 32; k > 0; k >>= 1) {
        if (t < k) sred[t] += sred[t + k];
        __syncthreads();
    }
    if (t == 0) { gm[0] = g; gm[1] = 1.0f / sred[0]; }
}

// Phase C3: out[i] = exp(e[i]-gmax) * invSum.  grid=128, block=256
__global__ void softmax_norm(const float* __restrict__ e,
                             const float* __restrict__ gm,
                             float* __restrict__ out) {
    const int i = blockIdx.x * 256 + threadIdx.x;
    out[i] = __expf(e[i] - gm[0]) * gm[1];
}

// ---------------------------------------------------------------------------
extern "C" void kernel_launch(void* const* d_in, const int* in_sizes, int n_in,
                              void* d_out, int out_size, void* d_ws, size_t ws_size,
                              hipStream_t stream) {
    const float* hidden = (const float*)d_in[0];   // [1024]
    const float* enc    = (const float*)d_in[1];   // [32768,1024]
    const float* W      = (const float*)d_in[2];   // [1024,1024]
    // d_in[3] = b is unused: softmax is invariant to the constant shift b.hidden

    float* ws      = (float*)d_ws;
    float* partial = ws;                 // 8*1024
    float* v       = ws + 8 * HIDDEN;    // 1024
    float* energ   = ws + 9 * HIDDEN;    // 32768
    float* stats   = energ + SEQ;        // 128
    float* gm      = stats + 128;        // 2
    float* out     = (float*)d_out;

    matvec_partial<<<dim3(4, 8), 256, 0, stream>>>(W, hidden, partial);
    matvec_reduce <<<4,          256, 0, stream>>>(partial, v);
    energies_wmma <<<SEQ / 128,  256, 0, stream>>>(enc, v, energ);
    softmax_part  <<<64,         256, 0, stream>>>(energ, stats);
    softmax_combine<<<1,          64, 0, stream>>>(stats, gm);
    softmax_norm  <<<SEQ / 256,  256, 0, stream>>>(energ, gm, out);
}